// QSANN_43774306681122
// MI455X (gfx1250) — compile-verified
//
#include <hip/hip_runtime.h>
#include <math.h>

// ---------------- problem constants (from reference) ----------------
#define NQB   10          // qubits
#define LTOK  128         // tokens per text
#define BATCH 16          // texts
#define NTOK  (BATCH * LTOK)   // 2048
#define MFEAT 100         // expvals per qnode / feature dim
#define NPAD  112         // MFEAT padded to 7 N-tiles of 16

typedef __attribute__((ext_vector_type(2))) float v2f;
typedef __attribute__((ext_vector_type(8))) float v8f;

// ---------------- wave32 helpers ----------------
__device__ __forceinline__ float2 shflx2(float2 v, int m) {
  float2 r;
  r.x = __shfl_xor(v.x, m, 32);
  r.y = __shfl_xor(v.y, m, 32);
  return r;
}

__device__ __forceinline__ float wred(float v) {
  v += __shfl_xor(v, 16, 32);
  v += __shfl_xor(v, 8, 32);
  v += __shfl_xor(v, 4, 32);
  v += __shfl_xor(v, 2, 32);
  v += __shfl_xor(v, 1, 32);
  return v;
}

__device__ __forceinline__ float2 fsel(bool c, float2 a, float2 b) {
  float2 r; r.x = c ? a.x : b.x; r.y = c ? a.y : b.y; return r;
}

// ======================================================================
// Quantum state: 1024 complex amplitudes, one wave per qnode.
// Flat index = h*32 + lane  (h = register slot 0..31, lane = 0..31).
// Qubit q maps to bit (9-q):  q<=4 -> register bit (4-q); q>=5 -> lane bit (9-q).
// All gate helpers require q to be a compile-time constant (callers fully
// unroll their q loops) so st[] stays in VGPRs.
// ======================================================================

__device__ __forceinline__ void gate_rx(float2 st[32], int q, float th, int lane) {
  float s, c;
  __sincosf(0.5f * th, &s, &c);
  if (q <= 4) {
    const int dr = 1 << (4 - q);
    #pragma unroll
    for (int h = 0; h < 32; ++h) {
      if (h & dr) continue;
      float2 a0 = st[h], a1 = st[h | dr];
      // n0 = c*a0 - i s*a1 ; n1 = c*a1 - i s*a0
      st[h]      = make_float2(c * a0.x + s * a1.y, c * a0.y - s * a1.x);
      st[h | dr] = make_float2(c * a1.x + s * a0.y, c * a1.y - s * a0.x);
    }
  } else {
    const int ls = 1 << (9 - q);
    #pragma unroll
    for (int h = 0; h < 32; ++h) {
      float2 p = shflx2(st[h], ls);     // partner amplitude
      // symmetric for both halves: n = c*self - i s*partner
      st[h] = make_float2(c * st[h].x + s * p.y, c * st[h].y - s * p.x);
    }
  }
}

__device__ __forceinline__ void gate_ry(float2 st[32], int q, float th, int lane) {
  float s, c;
  __sincosf(0.5f * th, &s, &c);
  if (q <= 4) {
    const int dr = 1 << (4 - q);
    #pragma unroll
    for (int h = 0; h < 32; ++h) {
      if (h & dr) continue;
      float2 a0 = st[h], a1 = st[h | dr];
      st[h]      = make_float2(c * a0.x - s * a1.x, c * a0.y - s * a1.y);
      st[h | dr] = make_float2(c * a1.x + s * a0.x, c * a1.y + s * a0.y);
    }
  } else {
    const int ls = 1 << (9 - q);
    float sg = (lane & ls) ? s : -s;    // n0 = c a0 - s a1 ; n1 = c a1 + s a0
    #pragma unroll
    for (int h = 0; h < 32; ++h) {
      float2 p = shflx2(st[h], ls);
      st[h] = make_float2(c * st[h].x + sg * p.x, c * st[h].y + sg * p.y);
    }
  }
}

__device__ __forceinline__ void gate_rz(float2 st[32], int q, float th, int lane) {
  float s, c;
  __sincosf(0.5f * th, &s, &c);
  // bit clear: *(c - i s)  -> (c x + s y, c y - s x)
  // bit set  : *(c + i s)  -> (c x - s y, c y + s x)
  if (q <= 4) {
    const int dr = 1 << (4 - q);
    #pragma unroll
    for (int h = 0; h < 32; ++h) {
      float t = (h & dr) ? s : -s;
      float2 a = st[h];
      st[h] = make_float2(c * a.x - t * a.y, c * a.y + t * a.x);
    }
  } else {
    const int ls = 1 << (9 - q);
    float t = (lane & ls) ? s : -s;
    #pragma unroll
    for (int h = 0; h < 32; ++h) {
      float2 a = st[h];
      st[h] = make_float2(c * a.x - t * a.y, c * a.y + t * a.x);
    }
  }
}

__device__ __forceinline__ void gate_cnot(float2 st[32], int cq, int tq, int lane) {
  if (cq <= 4 && tq <= 4) {                       // both bits in registers
    const int dc = 1 << (4 - cq), dt = 1 << (4 - tq);
    #pragma unroll
    for (int h = 0; h < 32; ++h) {
      if ((h & dc) && !(h & dt)) {
        float2 t = st[h]; st[h] = st[h | dt]; st[h | dt] = t;
      }
    }
  } else if (cq <= 4) {                           // control in regs, target across lanes
    const int dc = 1 << (4 - cq), lt = 1 << (9 - tq);
    #pragma unroll
    for (int h = 0; h < 32; ++h)
      if (h & dc) st[h] = shflx2(st[h], lt);      // pure cross-lane swap
  } else if (tq <= 4) {                           // control across lanes, target in regs
    const int lc = 1 << (9 - cq), dt = 1 << (4 - tq);
    bool ctl = (lane & lc) != 0;
    #pragma unroll
    for (int h = 0; h < 32; ++h) {
      if (h & dt) continue;
      float2 a = st[h], b = st[h | dt];
      st[h]      = fsel(ctl, b, a);
      st[h | dt] = fsel(ctl, a, b);
    }
  } else {                                        // both bits across lanes
    const int lc = 1 << (9 - cq), lt = 1 << (9 - tq);
    bool ctl = (lane & lc) != 0;
    #pragma unroll
    for (int h = 0; h < 32; ++h) {
      float2 p = shflx2(st[h], lt);
      st[h] = fsel(ctl, p, st[h]);
    }
  }
}

// <Z>, 2*Re<conj(p0)p1>, 2*Im<conj(p0)p1> for qubit q (q compile-time const)
__device__ __forceinline__ void expval(const float2 st[32], int q, int lane,
                                       float& z, float& xo, float& yo) {
  float az = 0.f, ar = 0.f, ai = 0.f;
  if (q <= 4) {
    const int dr = 1 << (4 - q);
    #pragma unroll
    for (int h = 0; h < 32; ++h) {
      float p2 = st[h].x * st[h].x + st[h].y * st[h].y;
      az += (h & dr) ? -p2 : p2;
      if (!(h & dr)) {
        float2 a0 = st[h], a1 = st[h | dr];
        ar += a0.x * a1.x + a0.y * a1.y;   // Re(conj(a0)*a1)
        ai += a0.x * a1.y - a0.y * a1.x;   // Im(conj(a0)*a1)
      }
    }
  } else {
    const int ls = 1 << (9 - q);
    bool hi = (lane & ls) != 0;
    float sgn = hi ? -1.f : 1.f;
    float msk = hi ? 0.f : 1.f;            // only bit-clear lanes contribute to <a>
    #pragma unroll
    for (int h = 0; h < 32; ++h) {
      float p2 = st[h].x * st[h].x + st[h].y * st[h].y;
      az += sgn * p2;
      float2 p = shflx2(st[h], ls);
      ar += msk * (st[h].x * p.x + st[h].y * p.y);
      ai += msk * (st[h].x * p.y - st[h].y * p.x);
    }
  }
  z  = wred(az);
  xo = 2.f * wred(ar);
  yo = 2.f * wred(ai);
}

// ======================================================================
// Kernel 1: one wave per qnode evaluation.
// wave id -> (token, m) with m: 0=Q(z0 only), 1=K(z0 only), 2=V(100 expvals)
// __launch_bounds__(256, 1): allow the full VGPR budget so the 64-VGPR
// statevector never spills to scratch.
// ======================================================================
__global__ void __launch_bounds__(256, 1) qnode_kernel(
    const int* __restrict__ ids, const float* __restrict__ emb,
    const float* __restrict__ featbuf,   // stride NPAD, only first 100 used
    const float* __restrict__ qw, const float* __restrict__ kw,
    const float* __restrict__ vw, int layer,
    float* __restrict__ qvo, float* __restrict__ kvo, float* __restrict__ vvo) {
  int wid   = blockIdx.x * (blockDim.x >> 5) + (threadIdx.x >> 5);
  int lane  = threadIdx.x & 31;
  int m     = wid % 3;
  int token = wid / 3;
  if (token >= NTOK) return;

  // angle source: layer0 = gathered embedding rows, layer1 = previous feats
  const float* A = (layer == 0) ? (emb + 100ull * (unsigned)ids[token])
                                : (featbuf + (unsigned long long)NPAD * (unsigned)token);
  const float* W = ((m == 0) ? qw : (m == 1) ? kw : vw) + layer * 40;  // (2,10,2)

  float2 st[32];
  #pragma unroll
  for (int h = 0; h < 32; ++h) st[h] = make_float2(0.f, 0.f);
  st[0].x = (lane == 0) ? 1.f : 0.f;          // |0...0>

  // ---- embedding circuit ----
  #pragma unroll 1
  for (int d = 0; d < 2; ++d) {
    #pragma unroll
    for (int q = 0; q < NQB; ++q) {
      int q2  = (q + 1) % NQB;
      int idx = (d * NQB + q) * 2;            // row pair in emb
      gate_rx(st, q,  A[2 * idx + 0], lane);
      gate_rx(st, q2, A[2 * idx + 1], lane);
      gate_ry(st, q,  A[2 * idx + 2], lane);
      gate_ry(st, q2, A[2 * idx + 3], lane);
      gate_cnot(st, q, q2, lane);
    }
  }
  #pragma unroll
  for (int q = 0; q < NQB; ++q) {
    int idx = 40 + q;
    gate_rx(st, q, A[2 * idx + 0], lane);
    gate_ry(st, q, A[2 * idx + 1], lane);
  }

  // ---- variational circuit ----
  #pragma unroll 1
  for (int l = 0; l < 2; ++l) {
    #pragma unroll
    for (int q = 0; q < NQB; ++q) {
      gate_ry(st, q, W[l * 20 + q * 2 + 0], lane);
      gate_rz(st, q, W[l * 20 + q * 2 + 1], lane);
    }
    #pragma unroll
    for (int q = 0; q < NQB; ++q)
      gate_cnot(st, q, (q + 1) % NQB, lane);
  }

  if (m < 2) {
    // only expvals[0] = <Z_0>  (qubit 0 -> register bit 4)
    float acc = 0.f;
    #pragma unroll
    for (int h = 0; h < 32; ++h) {
      float p2 = st[h].x * st[h].x + st[h].y * st[h].y;
      acc += (h & 16) ? -p2 : p2;
    }
    float z0 = wred(acc);
    if (lane == 0) ((m == 0) ? qvo : kvo)[token] = z0;
  } else {
    // full 100-entry expval vector, tiled [z(10) y(10) x(10)] pattern.
    // After wred the values are wave-uniform; fan the 10 copies across
    // lanes 0..9 so each q needs a single static store instruction.
    float* o = vvo + 100ull * (unsigned)token;
    #pragma unroll
    for (int q = 0; q < NQB; ++q) {
      float z, xx, yy;
      expval(st, q, lane, z, xx, yy);
      if (lane < 10) {
        float val = (lane < 4) ? z : ((lane < 7) ? yy : xx);
        int   off = (lane < 4) ? (q + 30 * lane)
                  : (lane < 7) ? (10 + q + 30 * (lane - 4))
                               : (20 + q + 30 * (lane - 7));
        o[off] = val;
      }
    }
  }
}

// ======================================================================
// Kernel 2: attention mix per text using V_WMMA_F32_16X16X4_F32.
// feats[i][p] = (sum_j exp(-(kv[j]-qv[i])^2) * vv[j][p]) / (rowsum_i+eps) * pi
// Block = 256 threads (8 waves); wave w owns M-tile w (16 rows); loops 7
// N-tiles x 32 k-steps of 16x16x4 f32 WMMA. vv staged in LDS via float4,
// zero-padded to 112 columns; output written unconditionally at stride 112
// so no EXEC manipulation ever touches the WMMA loop.
// A layout (ISA 7.12.2): lane L holds M=L&15; VGPR0 = K {0|2}, VGPR1 = K {1|3}
// B layout (mirror): lane L holds N=L&15; VGPR0 = K-row {0|2}, VGPR1 = {1|3}
// D layout: VGPR r -> M = r + (lane>=16 ? 8 : 0), N = lane&15
// ======================================================================
__global__ void __launch_bounds__(256, 1) attn_kernel(
    const float* __restrict__ qv, const float* __restrict__ kv,
    const float* __restrict__ vv,        // stride MFEAT (=100)
    float* __restrict__ outf) {          // stride NPAD  (=112)
  __shared__ float qs[LTOK], ks[LTOK], rs[LTOK];
  __shared__ float vs[LTOK * NPAD];      // 56 KB

  int b = blockIdx.x, tid = threadIdx.x;
  if (tid < LTOK) {
    qs[tid] = qv[b * LTOK + tid];
    ks[tid] = kv[b * LTOK + tid];
  }
  __syncthreads();
  if (tid < LTOK) {
    float qi = qs[tid], s = 0.f;
    for (int j = 0; j < LTOK; ++j) {
      float d = ks[j] - qi;
      s += expf(-d * d);
    }
    rs[tid] = s;
  }
  // Stage vv (row-major, 100 f32/row, rows 16B aligned) into LDS padded to
  // 112 cols with zeros, using 128-bit transfers: 128 rows x 28 quads.
  {
    const float4* vv4 = (const float4*)(vv + (size_t)b * LTOK * MFEAT);
    for (int i = tid; i < LTOK * (NPAD / 4); i += 256) {
      int r = i / (NPAD / 4), cq = i % (NPAD / 4);
      float4 val = make_float4(0.f, 0.f, 0.f, 0.f);
      if (cq < MFEAT / 4) val = vv4[r * (MFEAT / 4) + cq];
      *(float4*)&vs[r * NPAD + cq * 4] = val;
    }
  }
  __syncthreads();

  int wave  = tid >> 5, lane = tid & 31;
  int mrow  = wave * 16 + (lane & 15);       // A-fragment row for this lane
  int khalf = (lane >> 4) * 2;               // K sub-offset: 0 (lo half) / 2 (hi half)
  float qi  = qs[mrow];

  // Precompute all A-fragments for this wave's M-tile: alpha(mrow, k)
  float a0[32], a1[32];
  #pragma unroll
  for (int kk = 0; kk < 32; ++kk) {
    int j0 = kk * 4 + khalf;
    float d0 = ks[j0] - qi, d1 = ks[j0 + 1] - qi;
    a0[kk] = expf(-d0 * d0);
    a1[kk] = expf(-d1 * d1);
  }

  const float PI_F = 3.14159265358979323846f;
  for (int nt = 0; nt < 7; ++nt) {
    int ncol = nt * 16 + (lane & 15);
    v8f acc = {};
    #pragma unroll
    for (int kk = 0; kk < 32; ++kk) {
      int k0 = kk * 4 + khalf;
      v2f Af; Af.x = a0[kk];                 Af.y = a1[kk];
      v2f Bf; Bf.x = vs[k0 * NPAD + ncol];   Bf.y = vs[(k0 + 1) * NPAD + ncol];
      // D = A(16x4) * B(4x16) + C   (f32 WMMA, wave32)
      acc = __builtin_amdgcn_wmma_f32_16x16x4_f32(
          /*neg_a=*/false, Af, /*neg_b=*/false, Bf,
          /*c_mod=*/(short)0, acc, /*reuse_a=*/false, /*reuse_b=*/false);
    }
    #pragma unroll
    for (int r = 0; r < 8; ++r) {
      int row = wave * 16 + r + ((lane >> 4) ? 8 : 0);
      float scale = PI_F / (rs[row] + 1e-9f);
      outf[((size_t)b * LTOK + row) * NPAD + ncol] = acc[r] * scale;
    }
  }
}

// ======================================================================
// Kernel 3: out[b] = sigmoid(mean_t(feats[b,t,:]) . weight + bias)
// feats at stride NPAD (only first 100 cols meaningful)
// ======================================================================
__global__ void __launch_bounds__(128, 1) head_kernel(
    const float* __restrict__ feats, const float* __restrict__ weight,
    const float* __restrict__ bias, float* __restrict__ out) {
  __shared__ float wsh[MFEAT];
  __shared__ float part[LTOK];
  int b = blockIdx.x, tid = threadIdx.x;
  if (tid < MFEAT) wsh[tid] = weight[tid];
  __syncthreads();

  const float* f = feats + ((size_t)b * LTOK + tid) * NPAD;
  float s = 0.f;
  #pragma unroll 4
  for (int p = 0; p < MFEAT; ++p) s += f[p] * wsh[p];
  part[tid] = s;
  __syncthreads();
  for (int off = 64; off >= 1; off >>= 1) {
    if (tid < off) part[tid] += part[tid + off];
    __syncthreads();
  }
  if (tid == 0) {
    float x = part[0] / (float)LTOK + bias[0];
    out[b] = 1.f / (1.f + expf(-x));
  }
}

// ======================================================================
// Launcher
// ======================================================================
extern "C" void kernel_launch(void* const* d_in, const int* in_sizes, int n_in,
                              void* d_out, int out_size, void* d_ws, size_t ws_size,
                              hipStream_t stream) {
  const int*   ids    = (const int*)d_in[0];    // (16,128) int32
  const float* emb    = (const float*)d_in[1];  // (8000,50,2)
  const float* weight = (const float*)d_in[2];  // (100,)
  const float* bias   = (const float*)d_in[3];  // (1,)
  const float* qw     = (const float*)d_in[4];  // (2,2,10,2)
  const float* kw     = (const float*)d_in[5];
  const float* vw     = (const float*)d_in[6];
  float* out = (float*)d_out;                   // (16,)

  float* ws  = (float*)d_ws;
  float* qvb = ws;                    // 2048
  float* kvb = qvb + NTOK;            // 2048
  float* vvb = kvb + NTOK;            // 2048*100
  float* ftb = vvb + NTOK * MFEAT;    // 2048*112 (padded feats, layer in/out)

  const int qnode_waves  = NTOK * 3;                 // 6144
  const int qnode_blocks = qnode_waves / 8;          // 256 thr = 8 waves/block

  // layer 0
  qnode_kernel<<<qnode_blocks, 256, 0, stream>>>(ids, emb, ftb, qw, kw, vw, 0,
                                                 qvb, kvb, vvb);
  attn_kernel<<<BATCH, 256, 0, stream>>>(qvb, kvb, vvb, ftb);
  // layer 1
  qnode_kernel<<<qnode_blocks, 256, 0, stream>>>(ids, emb, ftb, qw, kw, vw, 1,
                                                 qvb, kvb, vvb);
  attn_kernel<<<BATCH, 256, 0, stream>>>(qvb, kvb, vvb, ftb);
  // readout
  head_kernel<<<BATCH, 128, 0, stream>>>(ftb, weight, bias, out);
}